// RBFSoftmax_36258113913185
// MI455X (gfx1250) — compile-verified
//
#include <hip/hip_runtime.h>

typedef __attribute__((ext_vector_type(2))) float v2f;
typedef __attribute__((ext_vector_type(8))) float v8f;

// ---------------------------------------------------------------------------
// Prologue: row-wise sum of squares for x (B rows) and W (C rows).
// xsq[b] = sum_d x[b,d]^2 ; wsq[c] = sum_d W[c,d]^2
// ---------------------------------------------------------------------------
__global__ __launch_bounds__(256)
void rbf_sq_kernel(const float* __restrict__ x, const float* __restrict__ W,
                   float* __restrict__ xsq, float* __restrict__ wsq,
                   int Btot, int Ctot, int Dd) {
  int idx = blockIdx.x * blockDim.x + threadIdx.x;
  if (idx >= Btot + Ctot) return;
  const float* p = (idx < Btot) ? (x + (size_t)idx * Dd)
                                : (W + (size_t)(idx - Btot) * Dd);
  float s = 0.0f;
  for (int k = 0; k < Dd; k += 4) {
    float4 v = *reinterpret_cast<const float4*>(p + k);
    s = __builtin_fmaf(v.x, v.x, s);
    s = __builtin_fmaf(v.y, v.y, s);
    s = __builtin_fmaf(v.z, v.z, s);
    s = __builtin_fmaf(v.w, v.w, s);
  }
  if (idx < Btot) xsq[idx] = s;
  else            wsq[idx - Btot] = s;
}

// ---------------------------------------------------------------------------
// Main kernel: fp32 WMMA GEMM (xw = x . W^T) fused with the RBF epilogue.
//
// Per wave: 64x64 output tile as a 4x4 grid of 16x16 WMMA tiles.
// Per block (256 threads = 8 waves, 2x4 wave grid): 128(M) x 256(N) tile.
//
// V_WMMA_F32_16X16X4_F32 fragment layouts (ISA 7.12.2):
//   A (16x4, MxK): lane r=lane%16 -> M=r; lane half g=lane/16 selects
//                  K={0,1} (g=0) or K={2,3} (g=1); 2 VGPRs = float2.
//   B (4x16, KxN): mirrored -> lane r -> N=r, half g selects K pair.
//   Since W is [C,D] row-major, both fragments load with the same formula:
//     base + (r)*D + k0 + 2*g   (contiguous float2 -> global_load_b64)
//   C/D (16x16): VGPR i, lanes 0-15 -> M=i, lanes 16-31 -> M=i+8; N=lane%16.
// ---------------------------------------------------------------------------
__global__ __launch_bounds__(256)
void rbf_wmma_kernel(const float* __restrict__ x, const float* __restrict__ W,
                     const float* __restrict__ xsq, const float* __restrict__ wsq,
                     float* __restrict__ out, int Ctot, int Dd) {
  const int lane = threadIdx.x & 31;
  const int wave = threadIdx.x >> 5;
  const int r = lane & 15;   // row/col within 16-wide tile
  const int g = lane >> 4;   // K-pair selector (0 or 1)

  const int m_base = blockIdx.y * 128 + (wave >> 2) * 64;  // rows of x
  const int n_base = blockIdx.x * 256 + (wave & 3) * 64;   // rows of W (= cols)

  const float* pA[4];
  const float* pB[4];
#pragma unroll
  for (int i = 0; i < 4; ++i) {
    pA[i] = x + (size_t)(m_base + i * 16 + r) * Dd + 2 * g;
    pB[i] = W + (size_t)(n_base + i * 16 + r) * Dd + 2 * g;
  }

  v8f acc[4][4];
#pragma unroll
  for (int i = 0; i < 4; ++i)
#pragma unroll
    for (int j = 0; j < 4; ++j)
      acc[i][j] = v8f{};

  // K loop: 8 x global_load_b64 feed 16 x v_wmma_f32_16x16x4_f32 per step.
  for (int k = 0; k < Dd; k += 4) {
    v2f a[4], b[4];
#pragma unroll
    for (int i = 0; i < 4; ++i) a[i] = *reinterpret_cast<const v2f*>(pA[i] + k);
#pragma unroll
    for (int j = 0; j < 4; ++j) b[j] = *reinterpret_cast<const v2f*>(pB[j] + k);
#pragma unroll
    for (int i = 0; i < 4; ++i)
#pragma unroll
      for (int j = 0; j < 4; ++j)
        acc[i][j] = __builtin_amdgcn_wmma_f32_16x16x4_f32(
            /*neg_a=*/false, a[i], /*neg_b=*/false, b[j],
            /*c_mod=*/(short)0, acc[i][j],
            /*reuse_a=*/false, /*reuse_b=*/false);
  }

  // Epilogue: out[row,col] = 15 * exp(-(xsq[row] + wsq[col] - 2*xw))
#pragma unroll
  for (int j = 0; j < 4; ++j) {
    const int col = n_base + j * 16 + r;
    const float wq = wsq[col];
#pragma unroll
    for (int i = 0; i < 4; ++i) {
#pragma unroll
      for (int v = 0; v < 8; ++v) {
        const int row = m_base + i * 16 + v + 8 * g;
        const float xq = xsq[row];
        const float m = xq + wq - 2.0f * acc[i][j][v];
        out[(size_t)row * Ctot + col] = 15.0f * __expf(-m);
      }
    }
  }
}

// ---------------------------------------------------------------------------
// Launch
// ---------------------------------------------------------------------------
extern "C" void kernel_launch(void* const* d_in, const int* in_sizes, int n_in,
                              void* d_out, int out_size, void* d_ws, size_t ws_size,
                              hipStream_t stream) {
  const float* x = (const float*)d_in[0];   // [B, D] f32
  const float* W = (const float*)d_in[1];   // [C, D] f32
  float* out = (float*)d_out;               // [B, C] f32

  const int Dd   = 256;
  const int Btot = in_sizes[0] / Dd;        // 8192
  const int Ctot = in_sizes[1] / Dd;        // 4096

  float* xsq = (float*)d_ws;                // Btot floats
  float* wsq = xsq + Btot;                  // Ctot floats

  const int rows = Btot + Ctot;
  rbf_sq_kernel<<<(rows + 255) / 256, 256, 0, stream>>>(x, W, xsq, wsq,
                                                        Btot, Ctot, Dd);

  dim3 grid(Ctot / 256, Btot / 128);        // (16, 64)
  rbf_wmma_kernel<<<grid, 256, 0, stream>>>(x, W, xsq, wsq, out, Ctot, Dd);
}